// MPMModel_66460323938971
// MI455X (gfx1250) — compile-verified
//
#include <hip/hip_runtime.h>
#include <math.h>
#include <stdint.h>

#define N_GRID 128
#define GRID_NODES (N_GRID * N_GRID)

__device__ __constant__ const float kDX      = 1.0f / 128.0f;
__device__ __constant__ const float kINV_DX  = 128.0f;
__device__ __constant__ const float kDT      = 1.0e-4f;
__device__ __constant__ const float kP_VOL   = (1.0f / 256.0f) * (1.0f / 256.0f);
__device__ __constant__ const float kP_MASS  = (1.0f / 256.0f) * (1.0f / 256.0f);
__device__ __constant__ const float kMU_0    = 1000.0f / (2.0f * 1.2f);          // 416.666...
__device__ __constant__ const float kLAMBDA0 = 1000.0f * 0.2f / (1.2f * 0.6f);   // 277.777...

// ---- gfx1250 async / TDM feature detection -------------------------------
typedef int          v4i __attribute__((ext_vector_type(4)));
typedef int          v8i __attribute__((ext_vector_type(8)));
typedef unsigned int v4u __attribute__((ext_vector_type(4)));
typedef __attribute__((address_space(1))) v4i  as1_v4i;   // global
typedef __attribute__((address_space(3))) v4i  as3_v4i;   // LDS
typedef __attribute__((address_space(3))) void as3_void;  // LDS (for ptr->offset)

#if defined(__has_builtin)
#if __has_builtin(__builtin_amdgcn_tensor_load_to_lds)
#define HAVE_TDM 1
#endif
#if __has_builtin(__builtin_amdgcn_s_wait_tensorcnt)
#define HAVE_WAIT_TENSOR 1
#endif
#if __has_builtin(__builtin_amdgcn_global_load_async_to_lds_b128)
#define HAVE_ASYNC_LDS 1
#endif
#if __has_builtin(__builtin_amdgcn_s_wait_asynccnt)
#define HAVE_WAIT_ASYNC 1
#endif
#endif

// --------------------------------------------------------------------------
__global__ void zero_grid_kernel(float* __restrict__ g, int n) {
    int i = blockIdx.x * blockDim.x + threadIdx.x;
    if (i < n) g[i] = 0.0f;
}

// Quadratic B-spline weights for one dimension.
__device__ inline void bspline_w(float fx, float w[3]) {
    float t0 = 1.5f - fx;
    float t1 = fx - 1.0f;
    float t2 = fx - 0.5f;
    w[0] = 0.5f * t0 * t0;
    w[1] = 0.75f - t1 * t1;
    w[2] = 0.5f * t2 * t2;
}

// --------------------------------------------------------------------------
// P2G: full per-particle physics + scatter of (v contrib, mass) with the
// affine term folded in analytically:  w*(mom + A @ x_node).
__global__ void p2g_kernel(const float* __restrict__ x, const float* __restrict__ v,
                           const float* __restrict__ C, const float* __restrict__ F,
                           const int* __restrict__ mat, const float* __restrict__ Jp,
                           float* __restrict__ grid_v, float* __restrict__ grid_m,
                           float* __restrict__ F_out, int* __restrict__ mat_out,
                           float* __restrict__ Jp_out, int n) {
    int p = blockIdx.x * blockDim.x + threadIdx.x;
    if (p >= n) return;

    float x0 = x[2 * p + 0], x1 = x[2 * p + 1];
    float v0 = v[2 * p + 0], v1 = v[2 * p + 1];
    float C00 = C[4 * p + 0], C01 = C[4 * p + 1], C10 = C[4 * p + 2], C11 = C[4 * p + 3];
    float F00 = F[4 * p + 0], F01 = F[4 * p + 1], F10 = F[4 * p + 2], F11 = F[4 * p + 3];
    int   m  = mat[p];
    float jp = Jp[p];

    float gx0 = x0 * kINV_DX, gx1 = x1 * kINV_DX;
    int b0 = (int)floorf(gx0 - 0.5f);
    int b1 = (int)floorf(gx1 - 0.5f);
    float fx0 = gx0 - (float)b0;
    float fx1 = gx1 - (float)b1;
    float wx[3], wy[3];
    bspline_w(fx0, wx);
    bspline_w(fx1, wy);

    // F = F + DT * C @ F
    float nF00 = F00 + kDT * (C00 * F00 + C01 * F10);
    float nF01 = F01 + kDT * (C00 * F01 + C01 * F11);
    float nF10 = F10 + kDT * (C10 * F00 + C11 * F10);
    float nF11 = F11 + kDT * (C10 * F01 + C11 * F11);

    float h   = (m == 1) ? 0.3f : __expf(10.0f * (1.0f - jp));
    float mu  = (m == 0) ? 0.0f : kMU_0 * h;
    float lam = kLAMBDA0 * h;

    // Closed-form 2x2 SVD:  nF = U * diag(sx,sy) * Vt   (sy >= 0 after fix-up)
    float E  = 0.5f * (nF00 + nF11);
    float Fh = 0.5f * (nF00 - nF11);
    float G  = 0.5f * (nF10 + nF01);
    float H  = 0.5f * (nF10 - nF01);
    float Q  = sqrtf(E * E + H * H);
    float Rr = sqrtf(Fh * Fh + G * G);
    float sx = Q + Rr;
    float sy = Q - Rr;
    float a1 = atan2f(G, Fh);
    float a2 = atan2f(H, E);
    float th = 0.5f * (a2 - a1);
    float ph = 0.5f * (a2 + a1);
    float cph, sph, cth, sth;
    __sincosf(ph, &sph, &cph);
    __sincosf(th, &sth, &cth);
    float U00 = cph, U01 = -sph, U10 = sph, U11 = cph;
    float Vt00 = cth, Vt01 = -sth, Vt10 = sth, Vt11 = cth;
    if (sy < 0.0f) { sy = -sy; Vt10 = -Vt10; Vt11 = -Vt11; }

    float s0 = sx, s1 = sy;
    float c0 = fminf(fmaxf(s0, 1.0f - 0.025f), 1.0f + 0.0045f);
    float c1 = fminf(fmaxf(s1, 1.0f - 0.025f), 1.0f + 0.0045f);
    if (m == 2) {
        jp = jp * (s0 / c0) * (s1 / c1);
        s0 = c0; s1 = c1;
    }
    float J = s0 * s1;

    if (m == 0) {           // liquid: reset to sqrt(J)*I
        float sj = sqrtf(J);
        nF00 = sj; nF01 = 0.0f; nF10 = 0.0f; nF11 = sj;
    } else if (m == 2) {    // snow: reconstruct with clamped singular values
        nF00 = U00 * s0 * Vt00 + U01 * s1 * Vt10;
        nF01 = U00 * s0 * Vt01 + U01 * s1 * Vt11;
        nF10 = U10 * s0 * Vt00 + U11 * s1 * Vt10;
        nF11 = U10 * s0 * Vt01 + U11 * s1 * Vt11;
    }

    float R00 = U00 * Vt00 + U01 * Vt10, R01 = U00 * Vt01 + U01 * Vt11;
    float R10 = U10 * Vt00 + U11 * Vt10, R11 = U10 * Vt01 + U11 * Vt11;

    // stress = 2*mu*(F-R) @ F^T + I * lam*J*(J-1)
    float D00 = nF00 - R00, D01 = nF01 - R01, D10 = nF10 - R10, D11 = nF11 - R11;
    float diag = lam * J * (J - 1.0f);
    float S00 = 2.0f * mu * (D00 * nF00 + D01 * nF01) + diag;
    float S01 = 2.0f * mu * (D00 * nF10 + D01 * nF11);
    float S10 = 2.0f * mu * (D10 * nF00 + D11 * nF01);
    float S11 = 2.0f * mu * (D10 * nF10 + D11 * nF11) + diag;

    float sc = -kDT * kP_VOL * 4.0f * kINV_DX * kINV_DX;   // == -DT exactly
    float A00 = sc * S00 + kP_MASS * C00;
    float A01 = sc * S01 + kP_MASS * C01;
    float A10 = sc * S10 + kP_MASS * C10;
    float A11 = sc * S11 + kP_MASS * C11;

    float mom0 = kP_MASS * v0 - (A00 * x0 + A01 * x1);
    float mom1 = kP_MASS * v1 - (A10 * x0 + A11 * x1);

#pragma unroll
    for (int i = 0; i < 3; ++i) {
#pragma unroll
        for (int j = 0; j < 3; ++j) {
            int gi = b0 + i, gj = b1 + j;
            if ((unsigned)gi < (unsigned)N_GRID && (unsigned)gj < (unsigned)N_GRID) {
                float wt = wx[i] * wy[j];
                float nx = (float)gi * kDX;
                float ny = (float)gj * kDX;
                float cv0 = wt * (mom0 + A00 * nx + A01 * ny);
                float cv1 = wt * (mom1 + A10 * nx + A11 * ny);
                int idx = gi * N_GRID + gj;
                atomicAdd(&grid_v[2 * idx + 0], cv0);
                atomicAdd(&grid_v[2 * idx + 1], cv1);
                atomicAdd(&grid_m[idx], wt * kP_MASS);
            }
        }
    }

    F_out[4 * p + 0] = nF00;
    F_out[4 * p + 1] = nF01;
    F_out[4 * p + 2] = nF10;
    F_out[4 * p + 3] = nF11;
    mat_out[p] = m;
    Jp_out[p]  = jp;
}

// --------------------------------------------------------------------------
__global__ void grid_update_kernel(float* __restrict__ grid_v,
                                   const float* __restrict__ grid_m) {
    int nidx = blockIdx.x * blockDim.x + threadIdx.x;
    if (nidx >= GRID_NODES) return;
    int i = nidx / N_GRID;
    int j = nidx % N_GRID;
    float mass = grid_m[nidx];
    float vx = grid_v[2 * nidx + 0];
    float vy = grid_v[2 * nidx + 1];
    if (mass > 0.0f) {
        float inv = 1.0f / mass;
        vx *= inv;
        vy *= inv;
        vy += -kDT * 50.0f;
    }
    if (i < 3)          vx = fmaxf(vx, 0.0f);
    if (i >= N_GRID - 2) vx = fminf(vx, 0.0f);
    if (j < 3)          vy = fmaxf(vy, 0.0f);
    if (j >= N_GRID - 2) vy = fminf(vy, 0.0f);
    grid_v[2 * nidx + 0] = vx;
    grid_v[2 * nidx + 1] = vy;
}

// --------------------------------------------------------------------------
// G2P: stage full 128 KB grid_v into LDS — preferred path is one TDM
// TENSOR_LOAD_TO_LDS per workgroup (TENSORcnt), falling back to per-lane
// async global->LDS (ASYNCcnt), then gather 9 neighbors per particle.
__global__ void g2p_kernel(const float* __restrict__ x,
                           const float* __restrict__ grid_v_g,
                           float* __restrict__ x_out, float* __restrict__ v_out,
                           float* __restrict__ C_out, int n) {
    extern __shared__ float sgv[];   // GRID_NODES*2 floats = 131072 bytes

#if defined(HAVE_TDM)
    if (threadIdx.x == 0) {
        // D# for a 1D tile: 16384 elements x 8 bytes = 128 KB, contiguous.
        unsigned long long ga = (unsigned long long)(uintptr_t)grid_v_g;
        unsigned int lds_addr = (unsigned int)(uintptr_t)(as3_void*)sgv;
        v4u g0;
        g0.x = 1u;                                     // count=1 (valid), user mode
        g0.y = lds_addr;                               // lds_addr[31:0]
        g0.z = (unsigned int)ga;                       // global_addr[31:0]
        g0.w = ((unsigned int)(ga >> 32) & 0x01FFFFFFu) | (2u << 30);  // addr[56:32] | type=2
        v8i g1;
        g1[0] = 0x00030000;   // workgroup_mask=0 | data_size=3 (8B)
        g1[1] = 0x40000000;   // tensor_dim0[15:0]=16384 at bits[63:48]
        g1[2] = 0x00010000;   // tensor_dim0 hi=0 | tensor_dim1=1 at bits[95:80]
        g1[3] = 0x40000000;   // tensor_dim1 hi=0 | tile_dim0=16384 at bits[127:112]
        g1[4] = 0x00000001;   // tile_dim1=1 | tile_dim2=0
        g1[5] = 0x00004000;   // tensor_dim0_stride[31:0]=16384
        g1[6] = 0;            // stride hi | tensor_dim1_stride lo
        g1[7] = 0;
        v4i g2 = { 0, 0, 0, 0 };
        v4i g3 = { 0, 0, 0, 0 };
        v8i g4 = { 0, 0, 0, 0, 0, 0, 0, 0 };
        __builtin_amdgcn_tensor_load_to_lds(g0, g1, g2, g3, g4, 0);
    }
#if defined(HAVE_WAIT_TENSOR)
    __builtin_amdgcn_s_wait_tensorcnt(0);
#else
    asm volatile("s_wait_tensorcnt 0" ::: "memory");
#endif
#elif defined(HAVE_ASYNC_LDS)
    const int total_v4 = (GRID_NODES * 2) / 4;   // 8192 x 16B transfers
    for (int t = threadIdx.x; t < total_v4; t += blockDim.x) {
        __builtin_amdgcn_global_load_async_to_lds_b128(
            (as1_v4i*)(grid_v_g + 4 * t), (as3_v4i*)(sgv + 4 * t), 0, 0);
    }
#if defined(HAVE_WAIT_ASYNC)
    __builtin_amdgcn_s_wait_asynccnt(0);
#else
    asm volatile("s_wait_asynccnt 0" ::: "memory");
#endif
#else
    const int total_v4 = (GRID_NODES * 2) / 4;
    for (int t = threadIdx.x; t < total_v4; t += blockDim.x) {
        float4 d = ((const float4*)grid_v_g)[t];
        ((float4*)sgv)[t] = d;
    }
#endif
    __syncthreads();

    int p = blockIdx.x * blockDim.x + threadIdx.x;
    if (p >= n) return;

    float x0 = x[2 * p + 0], x1 = x[2 * p + 1];
    float gx0 = x0 * kINV_DX, gx1 = x1 * kINV_DX;
    int b0 = (int)floorf(gx0 - 0.5f);
    int b1 = (int)floorf(gx1 - 0.5f);
    float fx0 = gx0 - (float)b0;
    float fx1 = gx1 - (float)b1;
    float wx[3], wy[3];
    bspline_w(fx0, wx);
    bspline_w(fx1, wy);

    float nv0 = 0.0f, nv1 = 0.0f;
    float c00 = 0.0f, c01 = 0.0f, c10 = 0.0f, c11 = 0.0f;

#pragma unroll
    for (int i = 0; i < 3; ++i) {
#pragma unroll
        for (int j = 0; j < 3; ++j) {
            int gi = b0 + i, gj = b1 + j;
            int ci = min(max(gi, 0), N_GRID - 1);
            int cj = min(max(gj, 0), N_GRID - 1);
            int idx = ci * N_GRID + cj;
            float wt  = wx[i] * wy[j];
            float gv0 = sgv[2 * idx + 0];
            float gv1 = sgv[2 * idx + 1];
            float nx = (float)gi * kDX;
            float ny = (float)gj * kDX;
            nv0 += wt * gv0;
            nv1 += wt * gv1;
            c00 += wt * gv0 * nx;
            c01 += wt * gv0 * ny;
            c10 += wt * gv1 * nx;
            c11 += wt * gv1 * ny;
        }
    }

    float k = 4.0f * kINV_DX * kINV_DX;
    C_out[4 * p + 0] = (c00 - nv0 * x0) * k;
    C_out[4 * p + 1] = (c01 - nv0 * x1) * k;
    C_out[4 * p + 2] = (c10 - nv1 * x0) * k;
    C_out[4 * p + 3] = (c11 - nv1 * x1) * k;
    v_out[2 * p + 0] = nv0;
    v_out[2 * p + 1] = nv1;
    x_out[2 * p + 0] = x0 + kDT * nv0;
    x_out[2 * p + 1] = x1 + kDT * nv1;
}

// --------------------------------------------------------------------------
extern "C" void kernel_launch(void* const* d_in, const int* in_sizes, int n_in,
                              void* d_out, int out_size, void* d_ws, size_t ws_size,
                              hipStream_t stream) {
    const float* x   = (const float*)d_in[0];
    const float* v   = (const float*)d_in[1];
    const float* C   = (const float*)d_in[2];
    const float* F   = (const float*)d_in[3];
    const int*   mat = (const int*)d_in[4];
    const float* Jp  = (const float*)d_in[5];
    const int n = in_sizes[0] / 2;

    float* out    = (float*)d_out;
    float* out_x  = out;
    float* out_v  = out + 2 * (size_t)n;
    float* out_C  = out + 4 * (size_t)n;
    float* out_F  = out + 8 * (size_t)n;
    int*   out_m  = (int*)(out + 12 * (size_t)n);
    float* out_Jp = out + 13 * (size_t)n;

    float* grid_v = (float*)d_ws;                 // 2*GRID_NODES floats
    float* grid_m = grid_v + 2 * GRID_NODES;      // GRID_NODES floats

    const int grid_floats = 3 * GRID_NODES;
    zero_grid_kernel<<<(grid_floats + 255) / 256, 256, 0, stream>>>(grid_v, grid_floats);

    p2g_kernel<<<(n + 255) / 256, 256, 0, stream>>>(
        x, v, C, F, mat, Jp, grid_v, grid_m, out_F, out_m, out_Jp, n);

    grid_update_kernel<<<(GRID_NODES + 255) / 256, 256, 0, stream>>>(grid_v, grid_m);

    const size_t lds_bytes = (size_t)GRID_NODES * 2 * sizeof(float);  // 128 KB
    g2p_kernel<<<(n + 1023) / 1024, 1024, lds_bytes, stream>>>(
        x, grid_v, out_x, out_v, out_C, n);
}